// TransformerForMultiEntityPrediction_16269336118098
// MI455X (gfx1250) — compile-verified
//
#include <hip/hip_runtime.h>
#include <math.h>

typedef float v2f __attribute__((ext_vector_type(2)));
typedef float v8f __attribute__((ext_vector_type(8)));

#define BS   16
#define SEQL 4096
#define HID  768
#define NM   128
#define ED   256
#define NROWS (BS * NM)        /* 2048 */

/* workspace layout (bytes) */
#define OFF_MV      0u           /* 2048*768*4  = 6,291,456  */
#define OFF_LEMB    6291456u     /* 2048*256*4  = 2,097,152  */
#define OFF_LOGITS  8388608u     /* 2048*2048*4 = 16,777,216 */
#define OFF_ACC     25165824u    /* 2 floats + pad            */
#define OFF_EV      25166080u    /* 2048*256*4  = 2,097,152  */

/* ---- 1. mention mean vectors: one block per (b,m), direct span sum ---- */
__global__ void mention_mean_kernel(const float* __restrict__ seq,
                                    const int*   __restrict__ spans,
                                    float*       __restrict__ mv) {
    const int n  = blockIdx.x;            /* b*128 + m */
    const int b  = n >> 7;
    const int t  = threadIdx.x;           /* 0..255 -> 3 h-columns each */
    const int s0 = spans[2 * n + 0];
    const int s1 = spans[2 * n + 1];
    const float* base = seq + (size_t)b * SEQL * HID;
    float a0 = 0.f, a1 = 0.f, a2 = 0.f;
    for (int s = s0; s < s1; ++s) {
        const float* row = base + (size_t)s * HID;
        if (s + 1 < s1) __builtin_prefetch(row + HID + t, 0, 0);
        a0 += row[t];
        a1 += row[t + 256];
        a2 += row[t + 512];
    }
    const int len = s1 - s0;
    const float inv = (len > 0) ? (1.0f / (float)len) : 0.0f;  /* nan_to_num */
    float* out = mv + (size_t)n * HID;
    out[t]       = a0 * inv;
    out[t + 256] = a1 * inv;
    out[t + 512] = a2 * inv;
}

/* ---- 2. gather label embeddings: lemb[j][e] = table[labels[b,0,m]][e] ---- */
__global__ void gather_labels_kernel(const int*   __restrict__ labels,
                                     const float* __restrict__ table,
                                     float*       __restrict__ lemb) {
    const int idx = blockIdx.x * 256 + threadIdx.x;  /* 0..524287 */
    const int j = idx >> 8;
    const int e = idx & 255;
    const int b = j >> 7, m = j & 127;
    const int ent = labels[b * 256 + m];             /* labels[:,0] flat */
    lemb[idx] = table[(size_t)ent * ED + e];
}

/* ---- 3. zero loss accumulators ---- */
__global__ void init_acc_kernel(float* __restrict__ acc) {
    if (threadIdx.x < 2) acc[threadIdx.x] = 0.0f;
}

/* ---- 4. GEMM1: ev = mv @ W^T + bias, fp32 WMMA 16x16x4, wave/tile ---- */
__global__ void gemm1_wmma_kernel(const float* __restrict__ mv,
                                  const float* __restrict__ W,
                                  const float* __restrict__ bias,
                                  float* __restrict__ ev,
                                  float* __restrict__ ev_out) {
    const int lane = threadIdx.x & 31;
    const int wave = threadIdx.x >> 5;
    const int tile = blockIdx.x * 8 + wave;   /* 0..2047 */
    const int rowBase = (tile >> 4) * 16;     /* 128 row tiles */
    const int colBase = (tile & 15) * 16;     /* 16 col tiles  */
    const int half = lane >> 4;               /* 0: K0/K1, 1: K2/K3 */
    const int l15  = lane & 15;
    const int kOff = half * 2;
    const float* aPtr = mv + (size_t)(rowBase + l15) * HID + kOff;  /* A row-major */
    const float* bPtr = W  + (size_t)(colBase + l15) * HID + kOff;  /* B[k][n]=W[n][k] */
    v8f c = {0.f, 0.f, 0.f, 0.f, 0.f, 0.f, 0.f, 0.f};
    for (int k = 0; k < HID; k += 4) {
        v2f a  = *(const v2f*)(aPtr + k);
        v2f bb = *(const v2f*)(bPtr + k);
        c = __builtin_amdgcn_wmma_f32_16x16x4_f32(false, a, false, bb,
                                                  (short)0, c, false, false);
    }
    const float bval = bias[colBase + l15];   /* N = lane&15 for all C VGPRs */
#pragma unroll
    for (int i = 0; i < 8; ++i) {
        const int r = rowBase + i + half * 8; /* M = vgpr + 8*(lane>=16) */
        const int q = colBase + l15;
        const float v = c[i] + bval;
        ev[(size_t)r * ED + q]     = v;       /* aligned copy for GEMM2 */
        ev_out[(size_t)r * ED + q] = v;       /* entity_vectors output  */
    }
}

/* ---- 5. GEMM2: logits = ev @ lemb^T, fp32 WMMA 16x16x4, wave/tile ---- */
__global__ void gemm2_wmma_kernel(const float* __restrict__ ev,
                                  const float* __restrict__ lemb,
                                  float* __restrict__ logits) {
    const int lane = threadIdx.x & 31;
    const int wave = threadIdx.x >> 5;
    const int tile = blockIdx.x * 8 + wave;    /* 0..16383 */
    const int rowBase = (tile >> 7) * 16;      /* 128 row tiles */
    const int colBase = (tile & 127) * 16;     /* 128 col tiles */
    const int half = lane >> 4;
    const int l15  = lane & 15;
    const int kOff = half * 2;
    const float* aPtr = ev   + (size_t)(rowBase + l15) * ED + kOff;
    const float* bPtr = lemb + (size_t)(colBase + l15) * ED + kOff; /* B[k][n]=lemb[n][k] */
    v8f c = {0.f, 0.f, 0.f, 0.f, 0.f, 0.f, 0.f, 0.f};
    for (int k = 0; k < ED; k += 4) {
        v2f a  = *(const v2f*)(aPtr + k);
        v2f bb = *(const v2f*)(bPtr + k);
        c = __builtin_amdgcn_wmma_f32_16x16x4_f32(false, a, false, bb,
                                                  (short)0, c, false, false);
    }
#pragma unroll
    for (int i = 0; i < 8; ++i) {
        const int r = rowBase + i + half * 8;
        const int q = colBase + l15;
        logits[(size_t)r * NROWS + q] = c[i];
    }
}

/* ---- 6. per-row log-softmax diagonal NLL, masked accumulation ---- */
__global__ void row_loss_kernel(const float* __restrict__ logits,
                                const int*   __restrict__ labels,
                                float* __restrict__ acc) {
    __shared__ float red[256];
    const int i = blockIdx.x;
    const int t = threadIdx.x;
    const float* row = logits + (size_t)i * NROWS;
    float m = -INFINITY;
    for (int j = t; j < NROWS; j += 256) m = fmaxf(m, row[j]);
    red[t] = m; __syncthreads();
    for (int s = 128; s > 0; s >>= 1) {
        if (t < s) red[t] = fmaxf(red[t], red[t + s]);
        __syncthreads();
    }
    const float rowmax = red[0]; __syncthreads();
    float se = 0.f;
    for (int j = t; j < NROWS; j += 256) se += expf(row[j] - rowmax);
    red[t] = se; __syncthreads();
    for (int s = 128; s > 0; s >>= 1) {
        if (t < s) red[t] += red[t + s];
        __syncthreads();
    }
    if (t == 0) {
        const float lse     = rowmax + logf(red[0]);
        const float per_row = -(row[i] - lse);
        const int b = i >> 7, mm = i & 127;
        const int status = labels[b * 256 + 128 + mm];  /* labels[:,1] flat */
        const float valid = (status >= 0) ? 1.0f : 0.0f;
        atomicAdd(&acc[0], per_row * valid);
        atomicAdd(&acc[1], valid);
    }
}

/* ---- 7. finalize loss (nan_to_num semantics) ---- */
__global__ void finalize_kernel(const float* __restrict__ acc,
                                float* __restrict__ out) {
    float loss = (acc[1] > 0.0f) ? (acc[0] / acc[1]) : 0.0f;
    if (isnan(loss)) loss = 0.0f;
    out[0] = loss;
}

extern "C" void kernel_launch(void* const* d_in, const int* in_sizes, int n_in,
                              void* d_out, int out_size, void* d_ws, size_t ws_size,
                              hipStream_t stream) {
    const float* seq    = (const float*)d_in[0];  /* [16,4096,768] */
    const float* W      = (const float*)d_in[1];  /* [256,768]     */
    const float* bias   = (const float*)d_in[2];  /* [256]         */
    const float* table  = (const float*)d_in[3];  /* [50000,256]   */
    const int*   spans  = (const int*)d_in[4];    /* [16,128,2]    */
    const int*   labels = (const int*)d_in[5];    /* [16,2,128]    */
    float* out = (float*)d_out;                   /* [1 + 524288]  */

    char* ws = (char*)d_ws;
    float* mv     = (float*)(ws + OFF_MV);
    float* lemb   = (float*)(ws + OFF_LEMB);
    float* logits = (float*)(ws + OFF_LOGITS);
    float* acc    = (float*)(ws + OFF_ACC);
    float* ev     = (float*)(ws + OFF_EV);

    mention_mean_kernel<<<NROWS, 256, 0, stream>>>(seq, spans, mv);
    gather_labels_kernel<<<(NROWS * ED) / 256, 256, 0, stream>>>(labels, table, lemb);
    init_acc_kernel<<<1, 32, 0, stream>>>(acc);
    gemm1_wmma_kernel<<<(NROWS / 16) * (ED / 16) / 8, 256, 0, stream>>>(mv, W, bias, ev, out + 1);
    gemm2_wmma_kernel<<<(NROWS / 16) * (NROWS / 16) / 8, 256, 0, stream>>>(ev, lemb, logits);
    row_loss_kernel<<<NROWS, 256, 0, stream>>>(logits, labels, acc);
    finalize_kernel<<<1, 1, 0, stream>>>(acc, out);
}